// GCNLinkPrediction_50723563765964
// MI455X (gfx1250) — compile-verified
//
#include <hip/hip_runtime.h>
#include <hip/hip_bf16.h>

typedef float v2f __attribute__((ext_vector_type(2)));
typedef float v8f __attribute__((ext_vector_type(8)));

#define FDIM 64

// ---------------------------------------------------------------------------
// Degree accumulation: deg[dst] += 1 per edge
// ---------------------------------------------------------------------------
__global__ void deg_kernel(const long long* __restrict__ dst, float* __restrict__ deg, int E) {
    int i = blockIdx.x * blockDim.x + threadIdx.x;
    if (i < E) atomicAdd(&deg[(int)dst[i]], 1.0f);
}

// In-place: dinv = rsqrt(deg + 1)   (self-loop included)
__global__ void dinv_kernel(float* __restrict__ deg, int N) {
    int i = blockIdx.x * blockDim.x + threadIdx.x;
    if (i < N) deg[i] = rsqrtf(deg[i] + 1.0f);
}

// ---------------------------------------------------------------------------
// WMMA fp32 GEMM:  Y[N x 64] = X[N x 64] @ W[64 x 64]  (+bias) (+relu)
// One 128-thread block per 16-row tile; wave w computes cols [16w, 16w+16).
// Uses V_WMMA_F32_16X16X4_F32, K stepped by 4, 16 WMMA issues per tile.
// ---------------------------------------------------------------------------
__global__ __launch_bounds__(128) void gemm64_wmma(
    const float* __restrict__ X, const float* __restrict__ W,
    const float* __restrict__ bias, float* __restrict__ Y,
    int addBias, int doRelu)
{
    const int wave = threadIdx.x >> 5;     // 0..3 -> 16-col tile
    const int lane = threadIdx.x & 31;
    const int half = lane >> 4;            // 0: lanes 0-15, 1: lanes 16-31
    const int lm   = lane & 15;
    const int row0 = blockIdx.x * 16;
    const int ncol = wave * 16 + lm;

    const float* Xrow = X + (size_t)(row0 + lm) * FDIM;

    v8f acc = {};
#pragma unroll
    for (int k = 0; k < FDIM; k += 4) {
        // A tile 16x4: lanes 0-15 carry K = k+0,k+1 ; lanes 16-31 carry K = k+2,k+3
        const int ka = k + half * 2;
        v2f a = *(const v2f*)(Xrow + ka);
        // B tile 4x16: row K striped across lanes within a VGPR (mirror of A halves)
        v2f b;
        b.x = W[(size_t)(ka + 0) * FDIM + ncol];
        b.y = W[(size_t)(ka + 1) * FDIM + ncol];
        acc = __builtin_amdgcn_wmma_f32_16x16x4_f32(
            /*neg_a=*/false, a, /*neg_b=*/false, b,
            /*c_mod=*/(short)0, acc, /*reuse_a=*/false, /*reuse_b=*/false);
    }

    const float bv = addBias ? bias[ncol] : 0.0f;
#pragma unroll
    for (int v = 0; v < 8; ++v) {
        // C/D layout: lanes 0-15 -> M=v, lanes 16-31 -> M=v+8, N=lane%16
        const int m = v + half * 8;
        float val = acc[v] + bv;
        if (doRelu) val = fmaxf(val, 0.0f);
        Y[(size_t)(row0 + m) * FDIM + ncol] = val;
    }
}

// ---------------------------------------------------------------------------
// Edge aggregation: agg[dst] += h[src] * dinv[src]*dinv[dst]
// 16 consecutive lanes handle one edge (4 feats each, float4 gather -> 256B
// contiguous per edge, L2-resident), f32 atomics for the scatter.
// ---------------------------------------------------------------------------
__global__ void agg_kernel(const float* __restrict__ h,
                           const long long* __restrict__ src,
                           const long long* __restrict__ dst,
                           const float* __restrict__ dinv,
                           float* __restrict__ agg, int E)
{
    int tid = blockIdx.x * blockDim.x + threadIdx.x;   // E*16 work items
    if (tid >= E * 16) return;
    const int e = tid >> 4;
    const int q = (tid & 15) * 4;
    const int s = (int)src[e];
    const int d = (int)dst[e];
    const float norm = dinv[s] * dinv[d];
    const float4 hv = *(const float4*)(h + (size_t)s * FDIM + q);
    float* out = agg + (size_t)d * FDIM + q;
    atomicAdd(out + 0, hv.x * norm);
    atomicAdd(out + 1, hv.y * norm);
    atomicAdd(out + 2, hv.z * norm);
    atomicAdd(out + 3, hv.w * norm);
}

// ---------------------------------------------------------------------------
// out = relu(agg + h * dinv^2 + bias)     (self-loop term + bias + activation)
// ---------------------------------------------------------------------------
__global__ void combine_kernel(const float* __restrict__ agg,
                               const float* __restrict__ h,
                               const float* __restrict__ dinv,
                               const float* __restrict__ bias,
                               float* __restrict__ out, int N)
{
    int i = blockIdx.x * blockDim.x + threadIdx.x;
    if (i >= N * FDIM) return;
    const int n = i >> 6;
    const int f = i & 63;
    const float di = dinv[n];
    const float v = agg[i] + h[i] * di * di + bias[f];
    out[i] = fmaxf(v, 0.0f);
}

// ---------------------------------------------------------------------------
extern "C" void kernel_launch(void* const* d_in, const int* in_sizes, int n_in,
                              void* d_out, int out_size, void* d_ws, size_t ws_size,
                              hipStream_t stream)
{
    const float*      x    = (const float*)d_in[0];
    const long long*  eidx = (const long long*)d_in[1];
    const float* W1  = (const float*)d_in[2];  const float* b1  = (const float*)d_in[3];
    const float* W2  = (const float*)d_in[4];  const float* b2  = (const float*)d_in[5];
    const float* W3  = (const float*)d_in[6];  const float* b3  = (const float*)d_in[7];
    const float* fw1 = (const float*)d_in[8];  const float* fb1 = (const float*)d_in[9];
    const float* fw2 = (const float*)d_in[10]; const float* fb2 = (const float*)d_in[11];
    float* out = (float*)d_out;

    const int N = in_sizes[0] / FDIM;   // 100000
    const int E = in_sizes[1] / 2;      // 1000000
    const long long* src = eidx;
    const long long* dst = eidx + E;

    // workspace layout: dinv[N] | bufX[N*64] | bufH[N*64] | bufA[N*64]
    float* dinv = (float*)d_ws;
    float* bufX = dinv + N;
    float* bufH = bufX + (size_t)N * FDIM;
    float* bufA = bufH + (size_t)N * FDIM;

    const int TB = 256;
    const int gemmBlocks = N / 16;                    // N multiple of 16
    const int edgeBlocks = (E + TB - 1) / TB;
    const int aggBlocks  = (E * 16 + TB - 1) / TB;
    const int combBlocks = (N * FDIM + TB - 1) / TB;
    const int nodeBlocks = (N + TB - 1) / TB;

    // --- symmetric normalization: dinv = rsqrt(deg + 1) ---
    hipMemsetAsync(dinv, 0, sizeof(float) * (size_t)N, stream);
    deg_kernel<<<edgeBlocks, TB, 0, stream>>>(dst, dinv, E);
    dinv_kernel<<<nodeBlocks, TB, 0, stream>>>(dinv, N);

    // --- GCN layer 1 ---
    gemm64_wmma<<<gemmBlocks, 128, 0, stream>>>(x, W1, nullptr, bufH, 0, 0);
    hipMemsetAsync(bufA, 0, sizeof(float) * (size_t)N * FDIM, stream);
    agg_kernel<<<aggBlocks, TB, 0, stream>>>(bufH, src, dst, dinv, bufA, E);
    combine_kernel<<<combBlocks, TB, 0, stream>>>(bufA, bufH, dinv, b1, bufX, N);

    // --- GCN layer 2 ---
    gemm64_wmma<<<gemmBlocks, 128, 0, stream>>>(bufX, W2, nullptr, bufH, 0, 0);
    hipMemsetAsync(bufA, 0, sizeof(float) * (size_t)N * FDIM, stream);
    agg_kernel<<<aggBlocks, TB, 0, stream>>>(bufH, src, dst, dinv, bufA, E);
    combine_kernel<<<combBlocks, TB, 0, stream>>>(bufA, bufH, dinv, b2, bufX, N);

    // --- GCN layer 3 ---
    gemm64_wmma<<<gemmBlocks, 128, 0, stream>>>(bufX, W3, nullptr, bufH, 0, 0);
    hipMemsetAsync(bufA, 0, sizeof(float) * (size_t)N * FDIM, stream);
    agg_kernel<<<aggBlocks, TB, 0, stream>>>(bufH, src, dst, dinv, bufA, E);
    combine_kernel<<<combBlocks, TB, 0, stream>>>(bufA, bufH, dinv, b3, bufX, N);

    // --- MLP head ---
    gemm64_wmma<<<gemmBlocks, 128, 0, stream>>>(bufX, fw1, fb1, bufH, 1, 1);
    gemm64_wmma<<<gemmBlocks, 128, 0, stream>>>(bufH, fw2, fb2, out, 1, 0);
}